// LearnedGateDeltaModel_3204045603475
// MI455X (gfx1250) — compile-verified
//
#include <hip/hip_runtime.h>
#include <hip/hip_bf16.h>
#include <math.h>

// Problem constants (reference: B, L, H, V = 2048, 1024, 64, 64)
constexpr int kB  = 2048;
constexpr int kL  = 1024;
constexpr int kH  = 64;
constexpr int kV  = 64;
constexpr int kTC = 128;       // time-chunk length
constexpr int kNC = kL / kTC;  // 8 chunks

typedef float v2f __attribute__((ext_vector_type(2)));
typedef float v8f __attribute__((ext_vector_type(8)));

// D = A(16x4,f32) * B(4x16,f32) + C(16x16,f32), full fp32 WMMA on gfx1250.
__device__ __forceinline__ v8f wmma4(v2f a, v2f b, v8f c) {
  return __builtin_amdgcn_wmma_f32_16x16x4_f32(false, a, false, b, (short)0, c,
                                               false, false);
}

__device__ __forceinline__ float rlane(float v, int l) {
  return __builtin_bit_cast(
      float, __builtin_amdgcn_readlane(__builtin_bit_cast(int, v), l));
}

__device__ __forceinline__ float wred32(float x) {
#pragma unroll
  for (int m = 16; m >= 1; m >>= 1) x += __shfl_xor(x, m, 32);
  return x;
}

// ---------------------------------------------------------------------------
// Kernel 1: embed gather + FF (relu MLP 64->128->64) + residual + layernorm
// for one time-chunk. One wave per 16-token tile; both GEMMs via f32 WMMA.
// hbuf layout: [b][t_local (0..127)][h]
// ---------------------------------------------------------------------------
__global__ __launch_bounds__(32) void ff_ln_kernel(
    const int* __restrict__ seq, const float* __restrict__ embW,
    const float* __restrict__ w1, const float* __restrict__ b1,
    const float* __restrict__ w2, const float* __restrict__ b2,
    const float* __restrict__ lng, const float* __restrict__ lnb,
    float* __restrict__ hbuf, int chunk) {
  __shared__ float hT[16 * 68];   // embedding tile 16x64 (padded stride 68)
  __shared__ float fT[16 * 132];  // hidden tile 16x128 (padded stride 132)
  __shared__ float xT[16 * 68];   // residual tile 16x64

  const int lane = threadIdx.x;  // 0..31
  const int tile = blockIdx.x;   // 8 tiles per batch per chunk
  const int b    = tile >> 3;
  const int tt   = tile & 7;
  const int t0   = chunk * kTC + tt * 16;

  // ---- gather 16 embedding rows into LDS (ids are wave-uniform) ----
#pragma unroll 4
  for (int r = 0; r < 16; ++r) {
    int id = seq[(size_t)b * kL + t0 + r];
    hT[r * 68 + lane]      = embW[id * kH + lane];
    hT[r * 68 + lane + 32] = embW[id * kH + lane + 32];
  }
  __syncthreads();

  const int half = lane >> 4;  // K-half for A/B fragments
  const int nl   = lane & 15;  // row (A) / col (B,C) within tile

  // ---- GEMM1: (16x64) @ w1(64x128) + b1 ----
  v8f acc1[8];
#pragma unroll
  for (int nt = 0; nt < 8; ++nt) {
    float bias = b1[nt * 16 + nl];
#pragma unroll
    for (int r = 0; r < 8; ++r) acc1[nt][r] = bias;
  }
  for (int kk = 0; kk < 16; ++kk) {
    int kb = kk * 4 + half * 2;
    v2f a;
    a.x = hT[nl * 68 + kb];
    a.y = hT[nl * 68 + kb + 1];
#pragma unroll
    for (int nt = 0; nt < 8; ++nt) {
      v2f bb;
      bb.x = w1[kb * 128 + nt * 16 + nl];
      bb.y = w1[(kb + 1) * 128 + nt * 16 + nl];
      acc1[nt] = wmma4(a, bb, acc1[nt]);
    }
  }
  // relu -> LDS in row-major (C layout: reg r holds row r + 8*half)
#pragma unroll
  for (int nt = 0; nt < 8; ++nt)
#pragma unroll
    for (int r = 0; r < 8; ++r)
      fT[(r + 8 * half) * 132 + nt * 16 + nl] = fmaxf(acc1[nt][r], 0.0f);
  __syncthreads();

  // ---- GEMM2: (16x128) @ w2(128x64) + b2 ----
  v8f acc2[4];
#pragma unroll
  for (int nt = 0; nt < 4; ++nt) {
    float bias = b2[nt * 16 + nl];
#pragma unroll
    for (int r = 0; r < 8; ++r) acc2[nt][r] = bias;
  }
  for (int kk = 0; kk < 32; ++kk) {
    int kb = kk * 4 + half * 2;
    v2f a;
    a.x = fT[nl * 132 + kb];
    a.y = fT[nl * 132 + kb + 1];
#pragma unroll
    for (int nt = 0; nt < 4; ++nt) {
      v2f bb;
      bb.x = w2[kb * 64 + nt * 16 + nl];
      bb.y = w2[(kb + 1) * 64 + nt * 16 + nl];
      acc2[nt] = wmma4(a, bb, acc2[nt]);
    }
  }
  // ---- residual add ----
#pragma unroll
  for (int nt = 0; nt < 4; ++nt)
#pragma unroll
    for (int r = 0; r < 8; ++r) {
      int row = r + 8 * half, col = nt * 16 + nl;
      xT[row * 68 + col] = acc2[nt][r] + hT[row * 68 + col];
    }
  __syncthreads();

  // ---- layernorm: one token row per lane (lanes 0..15) ----
  if (lane < 16) {
    const int row = lane;
    float mu = 0.f;
#pragma unroll
    for (int j = 0; j < 64; ++j) mu += xT[row * 68 + j];
    mu *= (1.0f / 64.0f);
    float var = 0.f;
#pragma unroll
    for (int j = 0; j < 64; ++j) {
      float d = xT[row * 68 + j] - mu;
      var += d * d;
    }
    var *= (1.0f / 64.0f);
    float rs = rsqrtf(var + 1e-5f);
    size_t base = ((size_t)b * kTC + tt * 16 + row) * kH;
#pragma unroll
    for (int j = 0; j < 64; ++j)
      hbuf[base + j] = (xT[row * 68 + j] - mu) * rs * lng[j] + lnb[j];
  }
}

// ---------------------------------------------------------------------------
// One delta-rule step. Lane l owns M rows 2l (M0) and 2l+1 (M1).
// kv = (k[2l], k[2l+1]); k broadcast to SGPRs via v_readlane; all reductions
// use 4 independent accumulator chains for ILP / dual-issue.
// ---------------------------------------------------------------------------
__device__ __forceinline__ void delta_step(float (&M0)[64], float (&M1)[64],
                                           const float (&gcol)[64], float2 kv,
                                           float g2, float b1v, float b2v) {
  float sk[64];  // wave-uniform -> SGPRs
#pragma unroll
  for (int j = 0; j < 32; ++j) {
    sk[2 * j]     = rlane(kv.x, j);
    sk[2 * j + 1] = rlane(kv.y, j);
  }

  float inv = 1.0f / (wred32(kv.x * kv.x + kv.y * kv.y) + 1e-6f);

  // gate: sigmoid(relu(k @ gw1 + gb1) @ gw2 + gb2) -- lane m owns column m
  float h0 = 0.f, h1 = 0.f, h2 = 0.f, h3 = 0.f;
#pragma unroll
  for (int j = 0; j < 64; j += 4) {
    h0 = __builtin_fmaf(sk[j + 0], gcol[j + 0], h0);
    h1 = __builtin_fmaf(sk[j + 1], gcol[j + 1], h1);
    h2 = __builtin_fmaf(sk[j + 2], gcol[j + 2], h2);
    h3 = __builtin_fmaf(sk[j + 3], gcol[j + 3], h3);
  }
  float hm = ((h0 + h1) + (h2 + h3)) + b1v;
  float z  = wred32(fmaxf(hm, 0.f) * g2) + b2v;
  float g  = 1.0f / (1.0f + __expf(-z));

  // v_pred = M @ k : two rows per lane, 4 independent chains
  float r0a = 0.f, r0b = 0.f, r1a = 0.f, r1b = 0.f;
#pragma unroll
  for (int j = 0; j < 64; j += 2) {
    r0a = __builtin_fmaf(M0[j], sk[j], r0a);
    r1a = __builtin_fmaf(M1[j], sk[j], r1a);
    r0b = __builtin_fmaf(M0[j + 1], sk[j + 1], r0b);
    r1b = __builtin_fmaf(M1[j + 1], sk[j + 1], r1b);
  }
  float r0 = r0a + r0b;
  float r1 = r1a + r1b;

  // M += g * (k - v_pred/denom) (x) k   (all independent FMAs)
  float gd0 = g * (kv.x - r0 * inv);
  float gd1 = g * (kv.y - r1 * inv);
#pragma unroll
  for (int j = 0; j < 64; ++j) {
    M0[j] = __builtin_fmaf(gd0, sk[j], M0[j]);
    M1[j] = __builtin_fmaf(gd1, sk[j], M1[j]);
  }
}

// ---------------------------------------------------------------------------
// Kernel 2: gated delta-rule scan over one time-chunk. One wave32 per batch;
// M (64x64 f32) lives in 128 VGPRs. k loads are software-pipelined 4 deep
// (one global_load_b64 per step, issued 4 steps ahead). M persists in ws
// between chunk launches; last chunk emits ctx = M @ q.
// ---------------------------------------------------------------------------
__global__ __launch_bounds__(256, 1) void scan_kernel(
    const float* __restrict__ hbuf, const float* __restrict__ gw1,
    const float* __restrict__ gb1, const float* __restrict__ gw2,
    const float* __restrict__ gb2, float* __restrict__ Mbuf,
    float* __restrict__ ctx, int chunk) {
  const int lane = threadIdx.x & 31;
  const int wv   = threadIdx.x >> 5;
  const int b    = __builtin_amdgcn_readfirstlane((int)blockIdx.x * 8 + wv);

  float M0[64], M1[64];  // rows 2*lane and 2*lane+1
  float* __restrict__ mp = Mbuf + (size_t)b * (kH * kH) + (2 * lane) * kH;
  if (chunk == 0) {
#pragma unroll
    for (int j = 0; j < 64; ++j) { M0[j] = 0.f; M1[j] = 0.f; }
  } else {
#pragma unroll
    for (int j = 0; j < 64; ++j) M0[j] = mp[j];
#pragma unroll
    for (int j = 0; j < 64; ++j) M1[j] = mp[kH + j];
  }

  // Gate weights: lane m (duplicated in lanes 16..31, zeroed via g2) owns
  // column m of gate_w1.
  const int cm = lane & 15;
  float gcol[64];
#pragma unroll
  for (int j = 0; j < 64; ++j) gcol[j] = gw1[j * 16 + cm];
  const float g2  = (lane < 16) ? gw2[cm] : 0.f;
  const float b1v = gb1[cm];
  const float b2v = gb2[0];

  const float2* __restrict__ hb2 =
      (const float2*)(hbuf + (size_t)b * (kTC * kH));  // row pitch 32 float2
  const int nsteps = (chunk == kNC - 1) ? (kTC - 1) : kTC;  // 1023 total

  // depth-4 software pipeline on k
  float2 pk[4];
#pragma unroll
  for (int i = 0; i < 4; ++i) pk[i] = hb2[i * 32 + lane];

  const int main_steps = nsteps & ~3;  // 128 or 124
  for (int tb = 0; tb < main_steps; tb += 4) {
#pragma unroll
    for (int i = 0; i < 4; ++i) {
      float2 kv = pk[i];
      int tn = tb + i + 4;
      if (tn > kTC - 1) tn = kTC - 1;  // clamp inside this batch's chunk
      pk[i] = hb2[tn * 32 + lane];     // prefetch 4 steps ahead
      delta_step(M0, M1, gcol, kv, g2, b1v, b2v);
    }
  }
#pragma unroll
  for (int i = 0; i < 3; ++i) {  // remainder (0 or 3 steps)
    if (main_steps + i < nsteps)
      delta_step(M0, M1, gcol, pk[i], g2, b1v, b2v);
  }

  if (chunk == kNC - 1) {
    // ctx = M @ q, q = h[:, L-1, :] (local row kTC-1)
    float2 qv = hb2[(kTC - 1) * 32 + lane];
    float sq[64];
#pragma unroll
    for (int j = 0; j < 32; ++j) {
      sq[2 * j]     = rlane(qv.x, j);
      sq[2 * j + 1] = rlane(qv.y, j);
    }
    float c0a = 0.f, c0b = 0.f, c1a = 0.f, c1b = 0.f;
#pragma unroll
    for (int j = 0; j < 64; j += 2) {
      c0a = __builtin_fmaf(M0[j], sq[j], c0a);
      c1a = __builtin_fmaf(M1[j], sq[j], c1a);
      c0b = __builtin_fmaf(M0[j + 1], sq[j + 1], c0b);
      c1b = __builtin_fmaf(M1[j + 1], sq[j + 1], c1b);
    }
    float2 cv;
    cv.x = c0a + c0b;
    cv.y = c1a + c1b;
    ((float2*)ctx)[(size_t)b * 32 + lane] = cv;  // rows 2l, 2l+1 adjacent
  } else {
#pragma unroll
    for (int j = 0; j < 64; ++j) mp[j] = M0[j];
#pragma unroll
    for (int j = 0; j < 64; ++j) mp[kH + j] = M1[j];
  }
}

// ---------------------------------------------------------------------------
// Kernel 3: logits = (ctx @ read_w + read_b) @ out_w + out_b
// ---------------------------------------------------------------------------
__global__ __launch_bounds__(64) void head_kernel(
    const float* __restrict__ ctx, const float* __restrict__ rw,
    const float* __restrict__ rb, const float* __restrict__ ow,
    const float* __restrict__ ob, float* __restrict__ logits) {
  __shared__ float c[64], t1[64];
  const int b = blockIdx.x, j = threadIdx.x;
  c[j] = ctx[(size_t)b * kH + j];
  __syncthreads();
  float a = rb[j];
#pragma unroll
  for (int i = 0; i < 64; ++i) a = __builtin_fmaf(c[i], rw[i * 64 + j], a);
  t1[j] = a;
  __syncthreads();
  float z = ob[j];
#pragma unroll
  for (int i = 0; i < 64; ++i) z = __builtin_fmaf(t1[i], ow[i * 64 + j], z);
  logits[(size_t)b * kV + j] = z;
}

// ---------------------------------------------------------------------------
extern "C" void kernel_launch(void* const* d_in, const int* in_sizes, int n_in,
                              void* d_out, int out_size, void* d_ws,
                              size_t ws_size, hipStream_t stream) {
  const int*   seq  = (const int*)d_in[0];
  const float* embW = (const float*)d_in[1];
  const float* w1   = (const float*)d_in[2];
  const float* b1   = (const float*)d_in[3];
  const float* w2   = (const float*)d_in[4];
  const float* b2   = (const float*)d_in[5];
  const float* lng  = (const float*)d_in[6];
  const float* lnb  = (const float*)d_in[7];
  const float* gw1  = (const float*)d_in[8];
  const float* gb1  = (const float*)d_in[9];
  const float* gw2  = (const float*)d_in[10];
  const float* gb2  = (const float*)d_in[11];
  const float* rw   = (const float*)d_in[12];
  const float* rb   = (const float*)d_in[13];
  const float* ow   = (const float*)d_in[14];
  const float* ob   = (const float*)d_in[15];
  float* logits = (float*)d_out;

  // workspace: h chunk (64 MB) | M state (32 MB) | ctx (0.5 MB) ~= 101 MB
  float* hbuf = (float*)d_ws;
  float* Mbuf = hbuf + (size_t)kB * kTC * kH;
  float* ctx  = Mbuf + (size_t)kB * kH * kH;

  for (int c = 0; c < kNC; ++c) {
    ff_ln_kernel<<<kB * (kTC / 16), 32, 0, stream>>>(seq, embW, w1, b1, w2, b2,
                                                     lng, lnb, hbuf, c);
    scan_kernel<<<kB / 8, 256, 0, stream>>>(hbuf, gw1, gb1, gw2, gb2, Mbuf,
                                            ctx, c);
  }
  head_kernel<<<kB, 64, 0, stream>>>(ctx, rw, rb, ow, ob, logits);
}